// Embedding_67070209294814
// MI455X (gfx1250) — compile-verified
//
#include <hip/hip_runtime.h>

typedef float v2f __attribute__((ext_vector_type(2)));
typedef float v8f __attribute__((ext_vector_type(8)));

#define SPLIT_LOC 32
#define SCALING   2.0f      // lora_alpha / r = 32/16
#define SEQ       512
#define DIM       4096
#define RANK      16
#define VOCAB     32000

// Block = 16 tokens x 64 embedding columns; 4 waves, each wave computes one
// 16(token) x 16(col) tile:  D = base_gather + [a1w0s | a2w1s](16x32) @ [B1;B2](32x16)
// via 8 chained V_WMMA_F32_16X16X4_F32 (K=4 each, fp32 exact).
__global__ __launch_bounds__(128) void moe_lora_embed_kernel(
    const int*   __restrict__ x,
    const float* __restrict__ emb,
    const float* __restrict__ A1, const float* __restrict__ B1,
    const float* __restrict__ A2, const float* __restrict__ B2,
    const float* __restrict__ Wr_img,  const float* __restrict__ br_img,
    const float* __restrict__ Wr_text, const float* __restrict__ br_text,
    float* __restrict__ out)
{
    __shared__ float lds_a[16 * 32];  // pre-scaled combined LoRA activations
    __shared__ float lds_w[16 * 2];   // router weights * SCALING
    __shared__ int   lds_x[16];       // token ids for this tile

    const int t0      = blockIdx.x * 16;      // token tile base (tile never crosses b)
    const int d_block = blockIdx.y * 64;      // column chunk base
    const int tid     = threadIdx.x;

    // ---- Pass 1: router softmax per token (2-way, over logits [0:2]) ----
    if (tid < 16) {
        const int t = t0 + tid;
        const int v = x[t];
        lds_x[tid]  = v;
        const int s = t & (SEQ - 1);          // position within sequence
        const float* Wr = (s < SPLIT_LOC) ? Wr_img : Wr_text;
        const float* br = (s < SPLIT_LOC) ? br_img : br_text;
        const float l0 = Wr[v]         + br[0];
        const float l1 = Wr[VOCAB + v] + br[1];
        const float m  = fmaxf(l0, l1);
        const float e0 = __expf(l0 - m);
        const float e1 = __expf(l1 - m);
        const float inv = SCALING / (e0 + e1);
        lds_w[tid * 2 + 0] = e0 * inv;
        lds_w[tid * 2 + 1] = e1 * inv;
    }
    __syncthreads();

    // ---- Pass 2: gather LoRA A columns, fold in router weight + scaling ----
    for (int idx = tid; idx < 16 * 32; idx += 128) {   // uniform: 4 iters, no divergence
        const int i = idx >> 5;     // token in tile
        const int j = idx & 31;     // combined rank index (0..15 -> e1, 16..31 -> e2)
        const int v = lds_x[i];
        float val;
        if (j < RANK) val = lds_w[i * 2 + 0] * A1[j * VOCAB + v];
        else          val = lds_w[i * 2 + 1] * A2[(j - RANK) * VOCAB + v];
        lds_a[idx] = val;
    }
    __syncthreads();

    // ---- Pass 3: WMMA accumulate on top of gathered base embedding ----
    const int wave = tid >> 5;
    const int lane = tid & 31;
    const int row  = lane & 15;     // A: M row / B,C,D: N column index
    const int hi   = lane >> 4;     // half-wave selector
    const int ncol = d_block + wave * 16 + row;   // this lane's output column

    // A fragments (16x4 f32 per step): lane<16 holds K=kb,kb+1; lane>=16 K=kb+2,kb+3
    v2f afrag[8];
#pragma unroll
    for (int kk = 0; kk < 8; ++kk) {
        const int kb = kk * 4 + hi * 2;
        afrag[kk].x = lds_a[row * 32 + kb];
        afrag[kk].y = lds_a[row * 32 + kb + 1];
    }

    // C init = base embedding gather, directly in WMMA C/D layout:
    // VGPR j holds (M = j + hi*8, N = ncol)
    v8f c;
#pragma unroll
    for (int j = 0; j < 8; ++j) {
        const int m = j + hi * 8;
        const long long roff = (long long)lds_x[m] * DIM;
        c[j] = emb[roff + ncol];
    }

    // 8 chained f32 WMMAs: C += A_comb(16x32) * B_comb(32x16)
    // B_comb[k, n] = (k<16 ? B1 : B2)[n, k&15], B matrices are [D, r] row-major.
#pragma unroll
    for (int kk = 0; kk < 8; ++kk) {
        const int kb = kk * 4 + hi * 2;
        const float* Bp = (kb < RANK) ? B1 : B2;
        const int jj = kb & (RANK - 1);
        v2f bfrag;
        bfrag.x = Bp[ncol * RANK + jj];
        bfrag.y = Bp[ncol * RANK + jj + 1];
        c = __builtin_amdgcn_wmma_f32_16x16x4_f32(
                /*neg_a=*/false, afrag[kk],
                /*neg_b=*/false, bfrag,
                /*c_mod=*/(short)0, c,
                /*reuse_a=*/false, /*reuse_b=*/false);
    }

    // ---- Store D fragment straight to out[t, d] ----
#pragma unroll
    for (int j = 0; j < 8; ++j) {
        const int m = j + hi * 8;
        const long long t = t0 + m;
        out[t * (long long)DIM + ncol] = c[j];
    }
}

extern "C" void kernel_launch(void* const* d_in, const int* in_sizes, int n_in,
                              void* d_out, int out_size, void* d_ws, size_t ws_size,
                              hipStream_t stream)
{
    (void)n_in; (void)d_ws; (void)ws_size; (void)out_size;
    const int*   x       = (const int*)  d_in[0];
    const float* emb     = (const float*)d_in[1];
    const float* A1      = (const float*)d_in[2];
    const float* B1      = (const float*)d_in[3];
    const float* A2      = (const float*)d_in[4];
    const float* B2      = (const float*)d_in[5];
    const float* Wr_img  = (const float*)d_in[6];
    const float* br_img  = (const float*)d_in[7];
    const float* Wr_text = (const float*)d_in[8];
    const float* br_text = (const float*)d_in[9];
    float* out = (float*)d_out;

    const int tokens = in_sizes[0];           // B*S = 4096
    dim3 grid(tokens / 16, DIM / 64);         // 256 x 64 blocks
    moe_lora_embed_kernel<<<grid, 128, 0, stream>>>(
        x, emb, A1, B1, A2, B2, Wr_img, br_img, Wr_text, br_text, out);
}